// RelevancePropagationConv2d_22076131901616
// MI455X (gfx1250) — compile-verified
//
#include <hip/hip_runtime.h>
#include <math.h>

// RelevancePropagationConv2d (theta=180deg) for MI455X / gfx1250, wave32.
// Collapsed to: 3 GEMMs of 1024x128x576 in f32 via V_WMMA_F32_16X16X4_F32
// + per-(patch,filter) scalar math + unfold/fold.
// gemm1 stages tiles with CDNA5 async global->LDS copies (ASYNCcnt).

typedef __attribute__((ext_vector_type(2))) float v2f;
typedef __attribute__((ext_vector_type(8))) float v8f;

#define EPS    1e-5f
#define COS_T  (-1.0f)
#define SIN_T  (1.2246468525851679e-16f)

// problem sizes (fixed by reference)
#define NB   4
#define NC   64
#define NH   16
#define NF   128
#define DIM  576      // 64*3*3
#define BS   1024     // 4*16*16 patches

__device__ __forceinline__ v8f wmma4(v2f a, v2f b, v8f c) {
  // D = A(16x4,f32) x B(4x16,f32) + C(16x16,f32)
  return __builtin_amdgcn_wmma_f32_16x16x4_f32(false, a, false, b, (short)0, c,
                                               false, false);
}

// CDNA5 async copy: 16 bytes global -> LDS per active lane, no VGPR data.
__device__ __forceinline__ void async_b128(unsigned int lds_addr,
                                           const float* gptr) {
  asm volatile("global_load_async_to_lds_b128 %0, %1, off" ::"v"(lds_addr),
               "v"((unsigned long long)(uintptr_t)gptr)
               : "memory");
}
__device__ __forceinline__ void wait_async0() {
  asm volatile("s_wait_asynccnt 0x0" ::: "memory");
}

// ---- reduction helper over 192-thread block -------------------------------
__device__ __forceinline__ float block_reduce192(float s, float* red) {
  const int tid = threadIdx.x;
  red[tid] = s;
  __syncthreads();
  if (tid < 64) red[tid] += red[tid + 64] + red[tid + 128];
  __syncthreads();
  if (tid < 16) red[tid] += red[tid + 16] + red[tid + 32] + red[tid + 48];
  __syncthreads();
  if (tid < 4) red[tid] += red[tid + 4] + red[tid + 8] + red[tid + 12];
  __syncthreads();
  if (tid == 0) red[0] = red[0] + red[1] + red[2] + red[3];
  __syncthreads();
  return red[0];
}

// ---- K0: per-filter norm, W^T (for GEMM1 B), W .* What (for GEMM2 B2) -----
__global__ void prep_w(const float* __restrict__ w, float* __restrict__ wn,
                       float* __restrict__ wt, float* __restrict__ wsq) {
  __shared__ float red[192];
  const int f = blockIdx.x;
  float s = 0.f;
  for (int d = threadIdx.x; d < DIM; d += 192) {
    float v = w[f * DIM + d];
    s += v * v;
  }
  float tot = block_reduce192(s, red);
  float wnr = sqrtf(tot);
  if (threadIdx.x == 0) wn[f] = wnr;
  float inv = 1.0f / (wnr + EPS);
  for (int d = threadIdx.x; d < DIM; d += 192) {
    float v = w[f * DIM + d];
    wt[d * NF + f] = v;              // B for GEMM1: (K=d) x (N=f)
    wsq[f * DIM + d] = v * v * inv;  // w .* w_hat
  }
}

// ---- K1: unfold x -> Xcol (1024x576), per-patch ||x|| ---------------------
__global__ void unfold_norm(const float* __restrict__ x,
                            float* __restrict__ xcol,
                            float* __restrict__ nx) {
  __shared__ float red[192];
  const int b = blockIdx.x;
  const int n = b >> 8, oh = (b >> 4) & 15, ow = b & 15;
  float s = 0.f;
  for (int d = threadIdx.x; d < DIM; d += 192) {
    int c = d / 9, k = d - c * 9, fh = k / 3, fw = k - fh * 3;
    int h = oh - 1 + fh, ww = ow - 1 + fw;
    float v = 0.f;
    if ((unsigned)h < 16u && (unsigned)ww < 16u)
      v = x[((n * NC + c) * NH + h) * NH + ww];
    xcol[b * DIM + d] = v;
    s += v * v;
  }
  float tot = block_reduce192(s, red);
  if (threadIdx.x == 0) nx[b] = sqrtf(tot);
}

// ---- K2: P = Xcol(1024x576) @ Wt(576x128), async-staged WMMA GEMM ---------
// block = 128 thr (4 waves), block tile 64M x 16N, wave tile 16x16, KC=32.
// Tiles staged via global_load_async_to_lds_b128 (no VGPR round-trip).
__global__ void gemm1(const float* __restrict__ xcol,
                      const float* __restrict__ wt, float* __restrict__ P) {
  __shared__ float Asub[64][36];  // 32 + 4 pad: rows 16B aligned, bank-spread
  __shared__ float Bsub[32][16];
  const int m0 = blockIdx.x * 64;
  const int n0 = blockIdx.y * 16;
  const int tid = threadIdx.x;
  const int wv = tid >> 5;          // wave 0..3 -> M sub-tile
  const int lane = tid & 31;
  const int l = lane & 15;
  const int kh = (lane >> 4) << 1;  // lanes 16..31 hold K+2,K+3
  const int mo = wv * 16;

  // per-thread async-copy targets
  const int arow = tid >> 3;       // 0..15 (4 row-groups of 16)
  const int acol = (tid & 7) * 4;  // 0,4,...,28
  const int brow = tid >> 2;       // 0..31
  const int bseg = (tid & 3) * 4;  // 0,4,8,12
  const unsigned int lds_a = (unsigned int)(uintptr_t)(&Asub[arow][acol]);
  const unsigned int lds_b = (unsigned int)(uintptr_t)(&Bsub[brow][bseg]);

  v8f acc = {};
  for (int kc = 0; kc < DIM; kc += 32) {
    // A tile 64x32: 4 x b128 per thread
    for (int i = 0; i < 4; ++i) {
      async_b128(lds_a + (unsigned int)(i * 16 * 36 * 4),
                 &xcol[(m0 + arow + i * 16) * DIM + kc + acol]);
    }
    // B tile 32x16: 1 x b128 per thread
    async_b128(lds_b, &wt[(kc + brow) * NF + n0 + bseg]);
    wait_async0();
    __syncthreads();
    for (int kk = 0; kk < 32; kk += 4) {
      v2f a, b;
      a.x = Asub[mo + l][kk + kh];
      a.y = Asub[mo + l][kk + kh + 1];
      b.x = Bsub[kk + kh][l];
      b.y = Bsub[kk + kh + 1][l];
      acc = wmma4(a, b, acc);
    }
    __syncthreads();
  }
  const int rbase = m0 + mo + ((lane >> 4) << 3);
  for (int i = 0; i < 8; ++i)
    P[(rbase + i) * NF + n0 + l] = acc[i];
}

// ---- K3: per-(patch,filter) scalar chain -> G1, G2 ------------------------
__global__ void scalars(const float* __restrict__ P,
                        const float* __restrict__ nxarr,
                        const float* __restrict__ wnarr,
                        const float* __restrict__ r, float* __restrict__ G1,
                        float* __restrict__ G2) {
  int idx = blockIdx.x * 256 + threadIdx.x;  // b*128+f
  int b = idx >> 7, f = idx & 127;
  int n = b >> 8, pix = b & 255;
  float p = P[idx];
  float nxr = nxarr[b];
  float nx = nxr + EPS;
  float wnr = wnarr[f];
  float wno = wnr + EPS;
  float rv = r[(n * NF + f) * 256 + pix];

  float wx = p / (nx * wno);              // xhat . what
  float wh2 = (wnr * wnr) / (wno * wno);  // ||what||^2
  float xh2 = (nxr * nxr) / (nx * nx);    // ||xhat||^2
  float o2 = wh2 - 2.f * wx * wx + wx * wx * xh2;
  float o = sqrtf(fmaxf(o2, 0.f)) + EPS;  // ||orth|| + eps
  float Bv = SIN_T * nx / o;
  float wdoto = (wnr * wnr / wno - wx * (p / nx)) / o;  // w . orth_hat
  float s = p * (1.f - COS_T) - nx * SIN_T * wdoto;     // sum_d w*x_diff
  float cc = rv / (s + EPS);
  G1[idx] = cc * ((1.f - COS_T) + Bv * wx / nx);
  G2[idx] = cc * Bv;
}

// ---- K4: fused dual GEMM: Rn = Xcol .* (G1@W) - (G2@Wsq) ------------------
// block = 128 thr (4 waves); each wave one 16x16 tile; grid (64 Mtiles, 9).
__global__ void gemm2(const float* __restrict__ G1,
                      const float* __restrict__ G2,
                      const float* __restrict__ w,
                      const float* __restrict__ wsq,
                      const float* __restrict__ xcol, float* __restrict__ Rn) {
  const int tid = threadIdx.x;
  const int wv = tid >> 5;
  const int lane = tid & 31;
  const int l = lane & 15;
  const int kh = (lane >> 4) << 1;
  const int m0 = blockIdx.x * 16;
  const int n0 = (blockIdx.y * 4 + wv) * 16;

  const float* g1row = G1 + (m0 + l) * NF;
  const float* g2row = G2 + (m0 + l) * NF;

  v8f acc1 = {};
  v8f acc2 = {};
  for (int k = 0; k < NF; k += 4) {
    v2f a1, a2, b1, b2;
    a1.x = g1row[k + kh];
    a1.y = g1row[k + kh + 1];
    a2.x = g2row[k + kh];
    a2.y = g2row[k + kh + 1];
    b1.x = w[(k + kh) * DIM + n0 + l];
    b1.y = w[(k + kh + 1) * DIM + n0 + l];
    b2.x = wsq[(k + kh) * DIM + n0 + l];
    b2.y = wsq[(k + kh + 1) * DIM + n0 + l];
    acc1 = wmma4(a1, b1, acc1);
    acc2 = wmma4(a2, b2, acc2);
  }
  const int rbase = m0 + ((lane >> 4) << 3);
  for (int i = 0; i < 8; ++i) {
    int off = (rbase + i) * DIM + n0 + l;
    Rn[off] = xcol[off] * acc1[i] - acc2[i];
  }
}

// ---- K5: fold (adjoint of unfold) as 9-tap gather -------------------------
__global__ void fold_k(const float* __restrict__ rn, float* __restrict__ out) {
  int idx = blockIdx.x * 256 + threadIdx.x;  // (n,c,h,w)
  int ww = idx & 15, h = (idx >> 4) & 15, c = (idx >> 8) & 63, n = idx >> 14;
  float acc = 0.f;
  for (int fh = 0; fh < 3; ++fh) {
    int oh = h + 1 - fh;
    if ((unsigned)oh >= 16u) continue;
    for (int fw = 0; fw < 3; ++fw) {
      int ow = ww + 1 - fw;
      if ((unsigned)ow >= 16u) continue;
      int b = (n * 16 + oh) * 16 + ow;
      int d = c * 9 + fh * 3 + fw;
      acc += rn[b * DIM + d];
    }
  }
  out[idx] = acc;
}

extern "C" void kernel_launch(void* const* d_in, const int* in_sizes, int n_in,
                              void* d_out, int out_size, void* d_ws,
                              size_t ws_size, hipStream_t stream) {
  const float* x = (const float*)d_in[0];  // (4,64,16,16)
  const float* r = (const float*)d_in[1];  // (4,128,16,16)
  const float* w = (const float*)d_in[2];  // (128,64,3,3) == (128,576)

  float* ws = (float*)d_ws;
  float* Xcol = ws;               // 1024*576
  float* Wt   = Xcol + BS * DIM;  // 576*128
  float* Wsq  = Wt + DIM * NF;    // 128*576
  float* wn   = Wsq + NF * DIM;   // 128
  float* nx   = wn + NF;          // 1024
  float* P    = nx + BS;          // 1024*128
  float* G1   = P + BS * NF;      // 1024*128
  float* G2   = G1 + BS * NF;     // 1024*128
  float* Rn   = G2 + BS * NF;     // 1024*576

  prep_w<<<NF, 192, 0, stream>>>(w, wn, Wt, Wsq);
  unfold_norm<<<BS, 192, 0, stream>>>(x, Xcol, nx);
  gemm1<<<dim3(BS / 64, NF / 16), 128, 0, stream>>>(Xcol, Wt, P);
  scalars<<<(BS * NF) / 256, 256, 0, stream>>>(P, nx, wn, r, G1, G2);
  gemm2<<<dim3(BS / 16, 9), 128, 0, stream>>>(G1, G2, w, Wsq, Xcol, Rn);
  fold_k<<<(NB * NC * NH * NH) / 256, 256, 0, stream>>>(Rn, (float*)d_out);
}